// xNN_13623636263323
// MI455X (gfx1250) — compile-verified
//
#include <hip/hip_runtime.h>
#include <math.h>

// Problem constants (from reference): B=64, NAP=128, NUSER=128, F=10, HID=100
#define NAPc 128
#define NUc  128
#define BFc  640            // B*F
#define HIDc 100
#define NMAT 16384          // 128*128
#define NSWEEP 10           // fixed cyclic-Jacobi sweeps (quadratic convergence)

typedef float v2f __attribute__((ext_vector_type(2)));
typedef float v8f __attribute__((ext_vector_type(8)));

// D = A(16x4,f32) * B(4x16,f32) + C(16x16,f32)  -> v_wmma_f32_16x16x4_f32
__device__ __forceinline__ v8f wmma4(v2f a, v2f b, v8f c) {
  return __builtin_amdgcn_wmma_f32_16x16x4_f32(false, a, false, b, (short)0, c,
                                               false, false);
}

// Async global->LDS 16B copy (per lane), tracked by ASYNCcnt.
// LDS offset = low 32 bits of the flat pointer.
__device__ __forceinline__ void async_g2l_b128(void* lds_ptr, const void* gptr) {
  unsigned l = (unsigned)(__SIZE_TYPE__)lds_ptr;
  asm volatile("global_load_async_to_lds_b128 %0, %1, off"
               :: "v"(l), "v"(gptr) : "memory");
}
__device__ __forceinline__ void wait_async0() {
  asm volatile("s_wait_asynccnt 0" ::: "memory");
}

// ---------------------------------------------------------------------------
// 1) Gather/transpose: Aw[m,k,i] = Ain[b,k,i,f], m = b*F + f
// ---------------------------------------------------------------------------
__global__ void k_pack(const float* __restrict__ Ain, float* __restrict__ Aw) {
  int idx = blockIdx.x * 256 + threadIdx.x;
  if (idx >= BFc * NMAT) return;
  int i = idx & 127;
  int k = (idx >> 7) & 127;
  int m = idx >> 14;
  int b = m / 10, f = m - b * 10;
  Aw[idx] = Ain[(((b * NAPc + k) * NUc) + i) * 10 + f];
}

// ---------------------------------------------------------------------------
// 2) Gram: L[m] = A[m]^T A[m]. One block per m; A[m] staged into LDS with
//    GLOBAL_LOAD_ASYNC_TO_LDS_B128. Each wave owns one 16-row block (i0),
//    register-caches the A tile once, and computes TWO j-tiles at a time
//    (independent accumulator chains hide the per-WMMA LDS wait).
// ---------------------------------------------------------------------------
__global__ void k_gram(const float* __restrict__ Aw, float* __restrict__ L) {
  __shared__ float Am[NMAT];
  int m = blockIdx.x, tid = threadIdx.x;
  {
    const char* src = (const char*)(Aw + (size_t)m * NMAT);
    char* dst = (char*)&Am[0];
    for (int c = tid; c < NMAT / 4; c += 256)      // 4096 x 16B chunks
      async_g2l_b128(dst + c * 16, src + c * 16);
    wait_async0();
  }
  __syncthreads();

  int wave = tid >> 5, lane = tid & 31;
  int r = lane & 15, kk = (lane >> 4) << 1;        // ISA A/B operand striping
  int i0 = wave << 4;
  float* out = L + (size_t)m * NMAT;

  v2f areg[32];
#pragma unroll
  for (int kq = 0; kq < 32; ++kq) {                // (A^T)[i,k] = A[k,i]
    areg[kq].x = Am[(4 * kq + kk) * 128 + i0 + r];
    areg[kq].y = Am[(4 * kq + kk + 1) * 128 + i0 + r];
  }
  for (int j0 = 0; j0 < 128; j0 += 32) {
    v8f acc0 = {}, acc1 = {};
#pragma unroll
    for (int kq = 0; kq < 32; ++kq) {              // B[k,j] = A[k,j]
      v2f b0v, b1v;
      b0v.x = Am[(4 * kq + kk) * 128 + j0 + r];
      b0v.y = Am[(4 * kq + kk + 1) * 128 + j0 + r];
      b1v.x = Am[(4 * kq + kk) * 128 + j0 + 16 + r];
      b1v.y = Am[(4 * kq + kk + 1) * 128 + j0 + 16 + r];
      acc0 = wmma4(areg[kq], b0v, acc0);
      acc1 = wmma4(areg[kq], b1v, acc1);
    }
    int rbase = i0 + ((lane >> 4) << 3);
#pragma unroll
    for (int v = 0; v < 8; ++v) {
      out[(rbase + v) * 128 + j0 + r]      = acc0[v];
      out[(rbase + v) * 128 + j0 + 16 + r] = acc1[v];
    }
  }
}

// ---------------------------------------------------------------------------
// 3) Batched symmetric eigensolver: parallel cyclic Jacobi, one block per
//    matrix. A (64KB) + V (64KB) live in dynamic LDS (320KB/WGP budget);
//    A staged via async LDS loads. Round-robin ordering: 127 rounds of 64
//    disjoint (p,q) pairs. Then bitonic sort (ascending, to match eigh),
//    write sorted V, eigenvalues, and s[m,j] = sum_i V[i,j]/128 (vtx is
//    t-independent because xin is the constant 1/Nap).
// ---------------------------------------------------------------------------
__global__ void k_jacobi(const float* __restrict__ L, float* __restrict__ Vg,
                         float* __restrict__ evals, float* __restrict__ svec) {
  extern __shared__ float smem[];
  float* Am  = smem;                 // 16384
  float* Vm  = smem + NMAT;          // 16384
  float* cc  = smem + 2 * NMAT;      // 64
  float* ss  = cc + 64;              // 64
  int*   perm = (int*)(ss + 64);     // 128
  float* key = (float*)(perm + 128); // 128
  int*   col = (int*)(key + 128);    // 128

  int m = blockIdx.x, tid = threadIdx.x;
  {
    const char* src = (const char*)(L + (size_t)m * NMAT);
    char* dst = (char*)&Am[0];
    for (int c = tid; c < NMAT / 4; c += 256)
      async_g2l_b128(dst + c * 16, src + c * 16);
  }
  for (int idx = tid; idx < NMAT; idx += 256)
    Vm[idx] = ((idx >> 7) == (idx & 127)) ? 1.f : 0.f;
  if (tid < 128) perm[tid] = tid;
  wait_async0();
  __syncthreads();

  for (int sweep = 0; sweep < NSWEEP; ++sweep) {
    for (int round = 0; round < 127; ++round) {
      if (tid < 64) {                       // rotation angles for 64 pairs
        int p = perm[tid], q = perm[127 - tid];
        if (p > q) { int t = p; p = q; q = t; }
        float app = Am[p * 128 + p], aqq = Am[q * 128 + q],
              apq = Am[p * 128 + q];
        float c = 1.f, s = 0.f;
        if (fabsf(apq) > 1e-30f) {
          float tau = (aqq - app) / (2.f * apq);
          float t = copysignf(1.f, tau) / (fabsf(tau) + sqrtf(1.f + tau * tau));
          c = 1.f / sqrtf(1.f + t * t);
          s = t * c;
        }
        cc[tid] = c; ss[tid] = s;
      }
      __syncthreads();
      for (int idx = tid; idx < 64 * 128; idx += 256) {   // rows: A = J^T A
        int g = idx >> 7, k = idx & 127;
        int p = perm[g], q = perm[127 - g];
        if (p > q) { int t = p; p = q; q = t; }
        float c = cc[g], s = ss[g];
        float x = Am[p * 128 + k], y = Am[q * 128 + k];
        Am[p * 128 + k] = c * x - s * y;
        Am[q * 128 + k] = s * x + c * y;
      }
      __syncthreads();
      for (int idx = tid; idx < 64 * 128; idx += 256) {   // cols: A=AJ, V=VJ
        int g = idx >> 7, rr = idx & 127;
        int p = perm[g], q = perm[127 - g];
        if (p > q) { int t = p; p = q; q = t; }
        float c = cc[g], s = ss[g];
        float x = Am[rr * 128 + p], y = Am[rr * 128 + q];
        Am[rr * 128 + p] = c * x - s * y;
        Am[rr * 128 + q] = s * x + c * y;
        x = Vm[rr * 128 + p]; y = Vm[rr * 128 + q];
        Vm[rr * 128 + p] = c * x - s * y;
        Vm[rr * 128 + q] = s * x + c * y;
      }
      __syncthreads();
      int tv = 0;                                         // rotate ordering
      if (tid >= 1 && tid < 128) tv = perm[tid];
      __syncthreads();
      if (tid >= 1 && tid < 128) perm[tid == 127 ? 1 : tid + 1] = tv;
      __syncthreads();
    }
  }

  if (tid < 128) { key[tid] = Am[tid * 128 + tid]; col[tid] = tid; }
  __syncthreads();
  for (int ksz = 2; ksz <= 128; ksz <<= 1) {              // bitonic ascending
    for (int j = ksz >> 1; j > 0; j >>= 1) {
      if (tid < 128) {
        int ixj = tid ^ j;
        if (ixj > tid) {
          bool up = ((tid & ksz) == 0);
          float k0 = key[tid], k1 = key[ixj];
          if ((k0 > k1) == up) {
            key[tid] = k1; key[ixj] = k0;
            int c0 = col[tid]; col[tid] = col[ixj]; col[ixj] = c0;
          }
        }
      }
      __syncthreads();
    }
  }
  if (tid < 128) evals[m * 128 + tid] = key[tid];
  for (int idx = tid; idx < NMAT; idx += 256) {
    int i = idx >> 7, jn = idx & 127;
    Vg[(size_t)m * NMAT + idx] = Vm[i * 128 + col[jn]];
  }
  if (tid < 128) {
    int cs = col[tid];
    float sum = 0.f;
    for (int i = 0; i < 128; ++i) sum += Vm[i * 128 + cs];
    svec[m * 128 + tid] = sum * (1.f / 128.f);
  }
}

// ---------------------------------------------------------------------------
// 4) MLP layers 0/1 (tiny: 640x100, K=128/100) — VALU dot products.
// ---------------------------------------------------------------------------
__global__ void k_mlp0(const float* __restrict__ e, const float* __restrict__ W0,
                       const float* __restrict__ b0, float* __restrict__ h0) {
  int idx = blockIdx.x * 256 + threadIdx.x;
  if (idx >= BFc * HIDc) return;
  int rr = idx / HIDc, o = idx - rr * HIDc;
  float acc = b0[o];
  for (int k = 0; k < NUc; ++k) acc += e[rr * NUc + k] * W0[k * HIDc + o];
  h0[idx] = fmaxf(acc, 0.f);
}
__global__ void k_mlp1(const float* __restrict__ h0, const float* __restrict__ W1,
                       const float* __restrict__ b1, float* __restrict__ h1) {
  int idx = blockIdx.x * 256 + threadIdx.x;
  if (idx >= BFc * HIDc) return;
  int rr = idx / HIDc, o = idx - rr * HIDc;
  float acc = b1[o];
  for (int k = 0; k < HIDc; ++k) acc += h0[rr * HIDc + k] * W1[k * HIDc + o];
  h1[idx] = fmaxf(acc, 0.f);
}

// ---------------------------------------------------------------------------
// 5) MLP layer 2: Y[640,16384] = h1[640,100] @ W2[100,16384] + b2 (WMMA,
//    K=100 = 25 * 4). Each wave computes a 16x32 strip (two n-tiles sharing
//    the A registers -> two independent WMMA chains). 20480 strips, 2560 blocks.
// ---------------------------------------------------------------------------
__global__ void k_mlp2(const float* __restrict__ h1, const float* __restrict__ W2,
                       const float* __restrict__ b2, float* __restrict__ Y) {
  int wave = threadIdx.x >> 5, lane = threadIdx.x & 31;
  int dt = blockIdx.x * 8 + wave;          // double-tile id, 512 per row-block
  int m0 = (dt >> 9) << 4, n0 = (dt & 511) << 5;
  int r = lane & 15, kk = (lane >> 4) << 1;
  v2f areg[25];
#pragma unroll
  for (int kq = 0; kq < 25; ++kq)
    areg[kq] = *(const v2f*)&h1[(m0 + r) * HIDc + 4 * kq + kk];
  v8f acc0 = {}, acc1 = {};
#pragma unroll
  for (int kq = 0; kq < 25; ++kq) {
    v2f b0v, b1v;
    b0v.x = W2[(4 * kq + kk) * (NAPc * NUc) + n0 + r];
    b0v.y = W2[(4 * kq + kk + 1) * (NAPc * NUc) + n0 + r];
    b1v.x = W2[(4 * kq + kk) * (NAPc * NUc) + n0 + 16 + r];
    b1v.y = W2[(4 * kq + kk + 1) * (NAPc * NUc) + n0 + 16 + r];
    acc0 = wmma4(areg[kq], b0v, acc0);
    acc1 = wmma4(areg[kq], b1v, acc1);
  }
  float bias0 = b2[n0 + r], bias1 = b2[n0 + 16 + r];
  int rbase = m0 + ((lane >> 4) << 3);
#pragma unroll
  for (int v = 0; v < 8; ++v) {
    Y[(size_t)(rbase + v) * (NAPc * NUc) + n0 + r]      = acc0[v] + bias0;
    Y[(size_t)(rbase + v) * (NAPc * NUc) + n0 + 16 + r] = acc1[v] + bias1;
  }
}

// ---------------------------------------------------------------------------
// 6) Spectral reconstruction + exp: per m, out[:,t] block is
//    V[m] @ ( d[t*640+m,:] * s[m,:] ).  Scaled-d tile staged in LDS with a
//    129-float pitch (129 mod 64 banks = 1 -> conflict-free strided reads);
//    A = V[m] rows via 8B global loads into registers; two t-tiles per wave
//    iteration (dual WMMA chains); expf fused into the store.
//    out_flat[(t*640+m)*128 + i].
// ---------------------------------------------------------------------------
__global__ void k_recon(const float* __restrict__ Vg, const float* __restrict__ Y,
                        const float* __restrict__ svec, float* __restrict__ out) {
  extern __shared__ float Ds[];                 // 128 * 129 floats
  int m = blockIdx.x, tid = threadIdx.x;
  for (int idx = tid; idx < NMAT; idx += 256) {
    int t = idx >> 7, j = idx & 127;
    Ds[t * 129 + j] = Y[((size_t)t * BFc + m) * NUc + j] * svec[m * NUc + j];
  }
  __syncthreads();
  int wave = tid >> 5, lane = tid & 31;
  int r = lane & 15, kk = (lane >> 4) << 1;
  const float* Vb = Vg + (size_t)m * NMAT;
  int i0 = wave << 4;

  v2f areg[32];
#pragma unroll
  for (int kq = 0; kq < 32; ++kq)               // A[i,j] = V[m][i,j]
    areg[kq] = *(const v2f*)&Vb[(i0 + r) * 128 + 4 * kq + kk];

  for (int t0 = 0; t0 < 128; t0 += 32) {
    v8f acc0 = {}, acc1 = {};
#pragma unroll
    for (int kq = 0; kq < 32; ++kq) {           // B[j,t] = d*s
      v2f b0v, b1v;
      b0v.x = Ds[(t0 + r) * 129 + 4 * kq + kk];
      b0v.y = Ds[(t0 + r) * 129 + 4 * kq + kk + 1];
      b1v.x = Ds[(t0 + 16 + r) * 129 + 4 * kq + kk];
      b1v.y = Ds[(t0 + 16 + r) * 129 + 4 * kq + kk + 1];
      acc0 = wmma4(areg[kq], b0v, acc0);
      acc1 = wmma4(areg[kq], b1v, acc1);
    }
    int ib = i0 + ((lane >> 4) << 3);
    size_t base0 = ((size_t)(t0 + r) * BFc + m) * NUc + ib;
    size_t base1 = ((size_t)(t0 + 16 + r) * BFc + m) * NUc + ib;
#pragma unroll
    for (int v = 0; v < 8; ++v) {
      out[base0 + v] = expf(acc0[v]);
      out[base1 + v] = expf(acc1[v]);
    }
  }
}

// ---------------------------------------------------------------------------
extern "C" void kernel_launch(void* const* d_in, const int* in_sizes, int n_in,
                              void* d_out, int out_size, void* d_ws, size_t ws_size,
                              hipStream_t stream) {
  const float* Ain = (const float*)d_in[0];
  const float* W0  = (const float*)d_in[1];
  const float* b0  = (const float*)d_in[2];
  const float* W1  = (const float*)d_in[3];
  const float* b1  = (const float*)d_in[4];
  const float* W2  = (const float*)d_in[5];
  const float* b2  = (const float*)d_in[6];
  float* out = (float*)d_out;

  char* ws = (char*)d_ws;
  size_t off = 0;
  float* Aw = (float*)(ws + off); off += (size_t)BFc * NMAT * 4;  // 41.9 MB
  float* Lm = (float*)(ws + off); off += (size_t)BFc * NMAT * 4;  // 41.9 MB
  float* Vg = (float*)(ws + off); off += (size_t)BFc * NMAT * 4;  // 41.9 MB
  float* ev = (float*)(ws + off); off += (size_t)BFc * NUc * 4;
  float* sv = (float*)(ws + off); off += (size_t)BFc * NUc * 4;
  float* h0 = (float*)(ws + off); off += (size_t)BFc * HIDc * 4;
  float* h1 = (float*)(ws + off); off += (size_t)BFc * HIDc * 4;
  float* Y  = Aw;   // Aw dead after k_gram; reuse for the [640,16384] MLP out

  k_pack<<<(BFc * NMAT + 255) / 256, 256, 0, stream>>>(Ain, Aw);
  k_gram<<<BFc, 256, 0, stream>>>(Aw, Lm);
  size_t jac_lds = (size_t)(2 * NMAT + 64 + 64 + 128 + 128 + 128) * 4; // ~130KB
  k_jacobi<<<BFc, 256, jac_lds, stream>>>(Lm, Vg, ev, sv);
  k_mlp0<<<(BFc * HIDc + 255) / 256, 256, 0, stream>>>(ev, W0, b0, h0);
  k_mlp1<<<(BFc * HIDc + 255) / 256, 256, 0, stream>>>(h0, W1, b1, h1);
  k_mlp2<<<2560, 256, 0, stream>>>(h1, W2, b2, Y);
  k_recon<<<BFc, 256, (size_t)128 * 129 * 4, stream>>>(Vg, Y, sv, out);
}